// DISTLoss_22041772163500
// MI455X (gfx1250) — compile-verified
//
#include <hip/hip_runtime.h>
#include <hip/hip_bf16.h>
#include <cstdint>

// ---------------------------------------------------------------------------
// DIST loss (B=4096, C=32000, tau=beta=gamma=1):
//   pass1 (row_pass):  online softmax per row -> row normalizers (M, 1/S) for
//                      s & t, inter-Pearson per row, CE per row.
//   pass2 (col_pass):  re-read tiles, accumulate 5 column stats per row-group
//                      using CDNA5 async global->LDS double buffering.
//   pass3 (class_pass): per-class Pearson from column stats.
//   pass4 (final_pass): deterministic scalar reduction.
// Memory floor: 2 x 1.05 GB reads / 23.3 TB/s ~= 90 us; all stats fit in
// ~5.2 MB of workspace. No atomics anywhere -> bitwise deterministic.
// ---------------------------------------------------------------------------

#define K_B 4096
#define K_C 32000
#define TILE_C 128          // columns per col_pass block
#define ROW_GROUPS 8        // gridDim.y of col_pass
#define ROWS_PER_GROUP (K_B / ROW_GROUPS)   // 512
#define ROWS_PER_ITER 8     // rows loaded per iteration per block
#define COL_ITERS (ROWS_PER_GROUP / ROWS_PER_ITER)  // 64
#define EPSF 1e-8f

// ---- CDNA5 async global->LDS (ASYNCcnt-tracked) ---------------------------
__device__ __forceinline__ void async_copy16(const float* gsrc, void* ldst) {
    // low 32 bits of a flat LDS pointer are the LDS byte offset
    unsigned l = (unsigned)(uintptr_t)ldst;
    asm volatile("global_load_async_to_lds_b128 %0, %1, off"
                 :: "v"(l), "v"(gsrc) : "memory");
}
#define ASYNC_WAIT(n) asm volatile("s_wait_asynccnt " #n ::: "memory")

// ===========================================================================
// Pass 1: one block per row. Online softmax for z_s, z_t with rescaled sums:
//   ss  = sum exp(zs - ms)        st  = sum exp(zt - mt)
//   ss2 = sum exp(zs - ms)^2      st2 = sum exp(zt - mt)^2
//   sst = sum exp(zs - ms)*exp(zt - mt)   (rescaled by BOTH max factors)
// Then: sum(ys*yt) = sst/(ss*st), sum(ys^2)=ss2/ss^2, sum(yt^2)=st2/st^2.
// Softmax row mean is exactly 1/C, so inter-Pearson needs only these sums.
// z_s[row][label] is a uniform address -> one scalar load, no per-element
// compare in the hot loop.
// ===========================================================================
__global__ void __launch_bounds__(256)
row_pass(const float* __restrict__ zs, const float* __restrict__ zt,
         const int* __restrict__ labels,
         float4* __restrict__ rowstats,
         float* __restrict__ inter_row, float* __restrict__ ce_row) {
    const int row = blockIdx.x;
    const int tid = threadIdx.x;
    const float* ps = zs + (size_t)row * K_C;
    const float* pt = zt + (size_t)row * K_C;
    const float zlab = ps[labels[row]];   // uniform -> scalar load

    float ms = -INFINITY, ss = 0.f, ss2 = 0.f;
    float mt = -INFINITY, st = 0.f, st2 = 0.f;
    float sst = 0.f;

    const float4* ps4 = (const float4*)ps;
    const float4* pt4 = (const float4*)pt;
    for (int idx = tid; idx < K_C / 4; idx += 256) {
        float4 a = ps4[idx];
        float4 b = pt4[idx];
        float av[4] = {a.x, a.y, a.z, a.w};
        float bv[4] = {b.x, b.y, b.z, b.w};
#pragma unroll
        for (int k = 0; k < 4; ++k) {
            const float z = av[k], w = bv[k];
            float es, et;
            if (z > ms) {                      // rescale s-sums (rare)
                float f = __expf(ms - z);
                ss *= f; ss2 *= f * f; sst *= f;
                ms = z; es = 1.0f;
            } else {
                es = __expf(z - ms);
            }
            ss += es; ss2 += es * es;
            if (w > mt) {                      // rescale t-sums (rare)
                float f = __expf(mt - w);
                st *= f; st2 *= f * f; sst *= f;
                mt = w; et = 1.0f;
            } else {
                et = __expf(w - mt);
            }
            st += et; st2 += et * et;
            sst += es * et;
        }
    }

    // ---- wave32 shuffle reduction (combine online-softmax partials) ----
#pragma unroll
    for (int off = 16; off > 0; off >>= 1) {
        float ms2  = __shfl_down(ms,  off, 32);
        float ss_2 = __shfl_down(ss,  off, 32);
        float ss22 = __shfl_down(ss2, off, 32);
        float mt2  = __shfl_down(mt,  off, 32);
        float st_2 = __shfl_down(st,  off, 32);
        float st22 = __shfl_down(st2, off, 32);
        float sst2 = __shfl_down(sst, off, 32);
        float M  = fmaxf(ms, ms2);
        float f1 = __expf(ms  - M), f2 = __expf(ms2 - M);
        float N  = fmaxf(mt, mt2);
        float g1 = __expf(mt  - N), g2 = __expf(mt2 - N);
        ss  = ss  * f1      + ss_2 * f2;
        ss2 = ss2 * f1 * f1 + ss22 * f2 * f2;
        st  = st  * g1      + st_2 * g2;
        st2 = st2 * g1 * g1 + st22 * g2 * g2;
        sst = sst * f1 * g1 + sst2 * f2 * g2;
        ms = M; mt = N;
    }

    // ---- cross-wave (8 waves) via LDS, fixed order ----
    __shared__ float red[8][7];
    const int wave = tid >> 5, lane = tid & 31;
    if (lane == 0) {
        red[wave][0] = ms;  red[wave][1] = ss;  red[wave][2] = ss2;
        red[wave][3] = mt;  red[wave][4] = st;  red[wave][5] = st2;
        red[wave][6] = sst;
    }
    __syncthreads();
    if (tid == 0) {
        for (int j = 1; j < 8; ++j) {
            float ms2 = red[j][0], ss_2 = red[j][1], ss22 = red[j][2];
            float mt2 = red[j][3], st_2 = red[j][4], st22 = red[j][5];
            float sst2 = red[j][6];
            float M  = fmaxf(ms, ms2);
            float f1 = __expf(ms - M), f2 = __expf(ms2 - M);
            float N  = fmaxf(mt, mt2);
            float g1 = __expf(mt - N), g2 = __expf(mt2 - N);
            ss  = ss  * f1      + ss_2 * f2;
            ss2 = ss2 * f1 * f1 + ss22 * f2 * f2;
            st  = st  * g1      + st_2 * g2;
            st2 = st2 * g1 * g1 + st22 * g2 * g2;
            sst = sst * f1 * g1 + sst2 * f2 * g2;
            ms = M; mt = N;
        }
        const float invC = 1.0f / (float)K_C;
        float sum_yy = sst / (ss * st);
        float sum_ss = ss2 / (ss * ss);
        float sum_tt = st2 / (st * st);
        float num = sum_yy - invC;
        float den = sqrtf(fmaxf(sum_ss - invC, 0.f)) *
                    sqrtf(fmaxf(sum_tt - invC, 0.f)) + EPSF;
        inter_row[row] = num / den;
        ce_row[row]    = ms + __logf(ss) - zlab;
        rowstats[row]  = make_float4(ms, 1.0f / ss, mt, 1.0f / st);
    }
}

// ===========================================================================
// Pass 2: grid (250 col tiles, 8 row groups), 256 threads.
// Lane layout: tx4 = tid&31 owns 4 consecutive columns; ty = tid>>5 walks rows.
// Streams 16B/lane/matrix per iteration through LDS with async double
// buffering (global_load_async_to_lds_b128 + s_wait_asynccnt). Each lane
// reads back only its own LDS slot, and async loads complete in order, so
// s_wait_asynccnt <= 2 (the pair just issued) is the only sync needed.
// ===========================================================================
__global__ void __launch_bounds__(256)
col_pass(const float* __restrict__ zs, const float* __restrict__ zt,
         const float4* __restrict__ rowstats, float* __restrict__ partial) {
    const int tid = threadIdx.x;
    const int tx4 = tid & 31;
    const int ty  = tid >> 5;
    const int ct  = blockIdx.x;
    const int rg  = blockIdx.y;
    const int c0  = ct * TILE_C + tx4 * 4;

    __shared__ __align__(16) float bufS[2][ROWS_PER_ITER][TILE_C];
    __shared__ __align__(16) float bufT[2][ROWS_PER_ITER][TILE_C];

    float acc[5][4];
#pragma unroll
    for (int s = 0; s < 5; ++s)
#pragma unroll
        for (int k = 0; k < 4; ++k) acc[s][k] = 0.f;

    auto issue = [&](int i) {
        const int b = i & 1;
        const int r = rg * ROWS_PER_GROUP + i * ROWS_PER_ITER + ty;
        async_copy16(zs + (size_t)r * K_C + c0, &bufS[b][ty][tx4 * 4]);
        async_copy16(zt + (size_t)r * K_C + c0, &bufT[b][ty][tx4 * 4]);
    };

    issue(0);
    for (int i = 0; i < COL_ITERS; ++i) {
        if (i + 1 < COL_ITERS) { issue(i + 1); ASYNC_WAIT(2); }
        else                   { ASYNC_WAIT(0); }
        const int b = i & 1;
        const int r = rg * ROWS_PER_GROUP + i * ROWS_PER_ITER + ty;
        const float4 rsn = rowstats[r];           // (Ms, 1/Ss, Mt, 1/St)
        const float* s4 = &bufS[b][ty][tx4 * 4];  // lane reads only its own slot
        const float* t4 = &bufT[b][ty][tx4 * 4];
#pragma unroll
        for (int k = 0; k < 4; ++k) {
            float ys = __expf(s4[k] - rsn.x) * rsn.y;
            float yt = __expf(t4[k] - rsn.z) * rsn.w;
            acc[0][k] += ys;
            acc[1][k] += yt;
            acc[2][k] += ys * ys;
            acc[3][k] += yt * yt;
            acc[4][k] += ys * yt;
        }
    }

    // cross-ty (8-way) reduction per column, fixed order -> deterministic
    __shared__ float red[256];
    for (int stat = 0; stat < 5; ++stat) {
#pragma unroll
        for (int k = 0; k < 4; ++k) {
            __syncthreads();
            red[tid] = acc[stat][k];
            __syncthreads();
            if (ty == 0) {
                float v = 0.f;
#pragma unroll
                for (int j = 0; j < 8; ++j) v += red[j * 32 + tx4];
                partial[(size_t)(rg * 5 + stat) * K_C + c0 + k] = v;
            }
        }
    }
}

// ===========================================================================
// Pass 3: one class per thread (125 blocks x 256 = 32000). Reduce the 8
// row-group partials in fixed order, compute intra-Pearson per class,
// block-tree-reduce, emit one partial per block.
// ===========================================================================
__global__ void __launch_bounds__(256)
class_pass(const float* __restrict__ partial, float* __restrict__ intra_blk) {
    const int c = blockIdx.x * 256 + threadIdx.x;
    float Ssum = 0.f, Tsum = 0.f, SS = 0.f, TT = 0.f, ST = 0.f;
    for (int rg = 0; rg < ROW_GROUPS; ++rg) {
        const float* base = partial + (size_t)(rg * 5) * K_C;
        Ssum += base[0 * (size_t)K_C + c];
        Tsum += base[1 * (size_t)K_C + c];
        SS   += base[2 * (size_t)K_C + c];
        TT   += base[3 * (size_t)K_C + c];
        ST   += base[4 * (size_t)K_C + c];
    }
    const float invB = 1.0f / (float)K_B;
    float msn = Ssum * invB, mtn = Tsum * invB;
    float num = ST - (float)K_B * msn * mtn;
    float vs = fmaxf(SS - (float)K_B * msn * msn, 0.f);
    float vt = fmaxf(TT - (float)K_B * mtn * mtn, 0.f);
    float pear = num / (sqrtf(vs) * sqrtf(vt) + EPSF);

    __shared__ float red[256];
    red[threadIdx.x] = pear;
    __syncthreads();
    for (int s = 128; s > 0; s >>= 1) {
        if (threadIdx.x < s) red[threadIdx.x] += red[threadIdx.x + s];
        __syncthreads();
    }
    if (threadIdx.x == 0) intra_blk[blockIdx.x] = red[0];
}

// ===========================================================================
// Pass 4: final scalar. out = CE_mean + (1 - inter_mean) + (1 - intra_mean).
// ===========================================================================
__global__ void __launch_bounds__(256)
final_pass(const float* __restrict__ inter_row, const float* __restrict__ ce_row,
           const float* __restrict__ intra_blk, float* __restrict__ out) {
    const int tid = threadIdx.x;
    float a = 0.f, b = 0.f, c = 0.f;
    for (int i = tid; i < K_B; i += 256) { a += inter_row[i]; b += ce_row[i]; }
    for (int i = tid; i < 125; i += 256) c += intra_blk[i];
    __shared__ float r0[256], r1[256], r2[256];
    r0[tid] = a; r1[tid] = b; r2[tid] = c;
    __syncthreads();
    for (int s = 128; s > 0; s >>= 1) {
        if (tid < s) { r0[tid] += r0[tid + s]; r1[tid] += r1[tid + s]; r2[tid] += r2[tid + s]; }
        __syncthreads();
    }
    if (tid == 0) {
        float inter_mean = r0[0] / (float)K_B;
        float hard       = r1[0] / (float)K_B;
        float intra_mean = r2[0] / (float)K_C;
        out[0] = hard + (1.0f - inter_mean) + (1.0f - intra_mean);
    }
}

// ===========================================================================
extern "C" void kernel_launch(void* const* d_in, const int* in_sizes, int n_in,
                              void* d_out, int out_size, void* d_ws, size_t ws_size,
                              hipStream_t stream) {
    const float* zs     = (const float*)d_in[0];
    const float* zt     = (const float*)d_in[1];
    const int*   labels = (const int*)d_in[2];
    float* out = (float*)d_out;
    float* ws  = (float*)d_ws;

    // workspace layout (floats):
    float4* rowstats  = (float4*)ws;          // 4096 * 4          = 16384
    float*  inter_row = ws + 16384;           // 4096
    float*  ce_row    = ws + 20480;           // 4096
    float*  intra_blk = ws + 24576;           // 125 (padded to 256)
    float*  partial   = ws + 24832;           // 8 * 5 * 32000 = 1,280,000
                                              // total ~5.22 MB

    row_pass  <<<dim3(K_B),                256, 0, stream>>>(zs, zt, labels, rowstats, inter_row, ce_row);
    col_pass  <<<dim3(K_C / TILE_C, ROW_GROUPS), 256, 0, stream>>>(zs, zt, rowstats, partial);
    class_pass<<<dim3(K_C / 256),          256, 0, stream>>>(partial, intra_blk);
    final_pass<<<dim3(1),                  256, 0, stream>>>(inter_row, ce_row, intra_blk, out);
}